// GlobalFilterLayer_81801947120202
// MI455X (gfx1250) — compile-verified
//
#include <hip/hip_runtime.h>
#include <math.h>

typedef float v2f __attribute__((ext_vector_type(2)));
typedef float v8f __attribute__((ext_vector_type(8)));

// cos(2*pi*t/16), sin(2*pi*t/16) for t = 0..15
__device__ __constant__ float COS16[16] = {
    1.0f,  0.92387953f,  0.70710678f,  0.38268343f,
    0.0f, -0.38268343f, -0.70710678f, -0.92387953f,
   -1.0f, -0.92387953f, -0.70710678f, -0.38268343f,
    0.0f,  0.38268343f,  0.70710678f,  0.92387953f};
__device__ __constant__ float SIN16[16] = {
    0.0f,  0.38268343f,  0.70710678f,  0.92387953f,
    1.0f,  0.92387953f,  0.70710678f,  0.38268343f,
    0.0f, -0.38268343f, -0.70710678f, -0.92387953f,
   -1.0f, -0.92387953f, -0.70710678f, -0.38268343f};

#define LDSTRIDE 20   // row stride (dwords): 16B-aligned rows, bank-conflict free
#define BUFDW (16 * LDSTRIDE)

__device__ __forceinline__ v8f wmma4(v2f a, v2f b, v8f c) {
    // D = A(16x4) * B(4x16) + C, fp32
    return __builtin_amdgcn_wmma_f32_16x16x4_f32(
        false, a, false, b, (short)0, c, false, false);
}

// DFT matrix F[m][n] = exp(-2*pi*i*m*n/16) is SYMMETRIC: same fragment values
// serve as A-operand (16x4 K-slice) and B-operand (4x16 K-slice).
//   A layout: lane l -> row = l&15,                 colK = 4*kb + v + 2*(l>>4)
//   B layout: lane l -> rowK = 4*kb + v + 2*(l>>4), col  = l&15
__device__ __forceinline__ void build_frags(int lane, const float* tab,
                                            float sgn, v2f out[4]) {
    int lo = lane & 15, hi = lane >> 4;
#pragma unroll
    for (int kb = 0; kb < 4; ++kb) {
        int k0 = 4 * kb + 2 * hi;
        v2f r;
        r.x = sgn * tab[(lo * (k0 + 0)) & 15];
        r.y = sgn * tab[(lo * (k0 + 1)) & 15];
        out[kb] = r;
    }
}

__device__ __forceinline__ v2f lds_bfrag(const float* buf, int kb, int lo, int hi) {
    v2f r;
    r.x = buf[(4 * kb + 0 + 2 * hi) * LDSTRIDE + lo];
    r.y = buf[(4 * kb + 1 + 2 * hi) * LDSTRIDE + lo];
    return r;
}
__device__ __forceinline__ v2f lds_afrag(const float* buf, int kb, int lo, int hi) {
    v2f r;
    r.x = buf[lo * LDSTRIDE + 4 * kb + 0 + 2 * hi];
    r.y = buf[lo * LDSTRIDE + 4 * kb + 1 + 2 * hi];
    return r;
}
__device__ __forceinline__ void store_cd(float* buf, v8f m, int lo, int hi) {
#pragma unroll
    for (int j = 0; j < 8; ++j)
        buf[(j + 8 * hi) * LDSTRIDE + lo] = m[j];
}

// Forward 2D FFT of the real matrix currently in bufA:  X = F * x * F.
__device__ __forceinline__ void fft2_fwd(float* bufA, float* bufB,
                                         const v2f fr[4], const v2f fi[4],
                                         const v2f mfi[4], int lo, int hi,
                                         v8f& xr, v8f& xi) {
    v8f a1r = {}, a1i = {};
#pragma unroll
    for (int kb = 0; kb < 4; ++kb) {
        v2f b = lds_bfrag(bufA, kb, lo, hi);
        a1r = wmma4(fr[kb], b, a1r);
        a1i = wmma4(fi[kb], b, a1i);
    }
    store_cd(bufA, a1r, lo, hi);
    store_cd(bufB, a1i, lo, hi);
    v8f r = {}, im = {};
#pragma unroll
    for (int kb = 0; kb < 4; ++kb) {
        v2f ar = lds_afrag(bufA, kb, lo, hi);
        v2f ai = lds_afrag(bufB, kb, lo, hi);
        r  = wmma4(ar, fr[kb],  r);
        r  = wmma4(ai, mfi[kb], r);
        im = wmma4(ar, fi[kb],  im);
        im = wmma4(ai, fr[kb],  im);
    }
    xr = r; xi = im;
}

// ---------------- Kernel 1: per-sample mean / rstd -------------------------
__global__ __launch_bounds__(256) void gf_stats(const float* __restrict__ x,
                                                float* __restrict__ meanv,
                                                float* __restrict__ rstdv,
                                                int n_per_sample) {
    __shared__ float s1[256];
    __shared__ float s2[256];
    int b = blockIdx.x, t = threadIdx.x;
    const float4* xb = (const float4*)(x + (size_t)b * n_per_sample);
    int n4 = n_per_sample >> 2;
    float s = 0.f, q = 0.f;
    for (int i = t; i < n4; i += 256) {
        float4 v = xb[i];
        s += v.x + v.y + v.z + v.w;
        q += v.x * v.x + v.y * v.y + v.z * v.z + v.w * v.w;
    }
    s1[t] = s; s2[t] = q;
    __syncthreads();
    for (int off = 128; off > 0; off >>= 1) {
        if (t < off) { s1[t] += s1[t + off]; s2[t] += s2[t + off]; }
        __syncthreads();
    }
    if (t == 0) {
        float inv = 1.0f / (float)n_per_sample;
        float m = s1[0] * inv;
        float var = s2[0] * inv - m * m;
        meanv[b] = m;
        rstdv[b] = rsqrtf(var + 1e-5f);
    }
}

// ---------------- Kernel 2: precompute fft2(filt) per channel --------------
// Stores H interleaved (re,im) in C/D-fragment order:
//   Hri[c*512 + (j*32 + lane)*2 + {0,1}]  -> main kernel loads b64, coalesced.
__global__ __launch_bounds__(256) void gf_filtfft(const float* __restrict__ filt,
                                                  float* __restrict__ Hri,
                                                  int nch) {
    __shared__ __align__(16) float lds[8 * 2 * BUFDW];
    int lane = threadIdx.x & 31;
    int wave = threadIdx.x >> 5;
    int lo = lane & 15, hi = lane >> 4;
    float* bufA = &lds[wave * 2 * BUFDW];
    float* bufB = bufA + BUFDW;

    v2f fr[4], fi[4], mfi[4];
    build_frags(lane, COS16,  1.0f, fr);
    build_frags(lane, SIN16, -1.0f, fi);   // Fi = -sin
    build_frags(lane, SIN16,  1.0f, mfi);  // -Fi

    // lane covers contiguous elements e = lane*8 + i -> row e>>4, col e&15
    int rmBase = 5 * (lane >> 1) + 2 * (lane & 1);   // float4 index, stride 20dw
    float4* bufA4 = (float4*)bufA;

    int gw = blockIdx.x * 8 + wave;
    int nw = gridDim.x * 8;
    for (int c = gw; c < nch; c += nw) {
        const float4* ft4 = (const float4*)(filt + (size_t)c * 256);
        bufA4[rmBase]     = ft4[lane * 2];
        bufA4[rmBase + 1] = ft4[lane * 2 + 1];
        v8f xr, xi;
        fft2_fwd(bufA, bufB, fr, fi, mfi, lo, hi, xr, xi);
        float2* h2 = (float2*)(Hri + (size_t)c * 512);
#pragma unroll
        for (int j = 0; j < 8; ++j) {
            float2 h; h.x = xr[j]; h.y = xi[j];
            h2[j * 32 + lane] = h;
        }
    }
}

// ---------------- Kernel 3: main fused LN + spectral filter ----------------
// Each wave owns ONE channel c and iterates over the batch, so gamma/beta
// (contiguous layout) and H (C/D-fragment layout) stay resident in VGPRs.
// Per-tile global traffic = read x + write |y| only (the 268 MB floor).
__global__ __launch_bounds__(256) void gf_main(const float* __restrict__ x,
                                               const float* __restrict__ gamma,
                                               const float* __restrict__ beta,
                                               const float* __restrict__ meanv,
                                               const float* __restrict__ rstdv,
                                               const float* __restrict__ Hri,
                                               float* __restrict__ out,
                                               int B, int C, int wavesPerCh) {
    __shared__ __align__(16) float lds[8 * 2 * BUFDW];
    int lane = threadIdx.x & 31;
    int wave = threadIdx.x >> 5;
    int lo = lane & 15, hi = lane >> 4;
    float* bufA = &lds[wave * 2 * BUFDW];
    float* bufB = bufA + BUFDW;
    float4* bufA4 = (float4*)bufA;

    v2f fr[4], fi[4], mfi[4];
    build_frags(lane, COS16,  1.0f, fr);
    build_frags(lane, SIN16, -1.0f, fi);
    build_frags(lane, SIN16,  1.0f, mfi);

    int gw = blockIdx.x * 8 + wave;           // nwaves == C * wavesPerCh
    int c = gw % C;
    int bslot = gw / C;
    int rmBase = 5 * (lane >> 1) + 2 * (lane & 1);

    // ---- per-channel residents: gamma/beta (contiguous), H (fragment order)
    const float4* g4 = (const float4*)(gamma + (size_t)c * 256);
    const float4* be4 = (const float4*)(beta + (size_t)c * 256);
    float4 ga0 = g4[lane * 2], ga1 = g4[lane * 2 + 1];
    float4 bt0 = be4[lane * 2], bt1 = be4[lane * 2 + 1];
    const float2* h2 = (const float2*)(Hri + (size_t)c * 512);
    float hre[8], him[8];
#pragma unroll
    for (int j = 0; j < 8; ++j) {
        float2 h = h2[j * 32 + lane];
        hre[j] = h.x; him[j] = h.y;
    }

    for (int b = bslot; b < B; b += wavesPerCh) {   // wave-uniform: EXEC all-1s
        size_t tile = (size_t)b * C + c;
        const float4* xt4 = (const float4*)(x + tile * 256);
        float mu = meanv[b], rs = rstdv[b];
        // coalesced b128 load + normalize + row-major b128 store to LDS
        float4 xa = xt4[lane * 2], xb = xt4[lane * 2 + 1];
        float4 r0, r1;
        r0.x = (xa.x - mu) * rs * ga0.x + bt0.x;
        r0.y = (xa.y - mu) * rs * ga0.y + bt0.y;
        r0.z = (xa.z - mu) * rs * ga0.z + bt0.z;
        r0.w = (xa.w - mu) * rs * ga0.w + bt0.w;
        r1.x = (xb.x - mu) * rs * ga1.x + bt1.x;
        r1.y = (xb.y - mu) * rs * ga1.y + bt1.y;
        r1.z = (xb.z - mu) * rs * ga1.z + bt1.z;
        r1.w = (xb.w - mu) * rs * ga1.w + bt1.w;
        bufA4[rmBase]     = r0;
        bufA4[rmBase + 1] = r1;

        // forward FFT: 24 WMMA
        v8f xr, xi;
        fft2_fwd(bufA, bufB, fr, fi, mfi, lo, hi, xr, xi);

        // pointwise spectral multiply (H resident in VGPRs)
        v8f zr, zi;
#pragma unroll
        for (int j = 0; j < 8; ++j) {
            zr[j] = hre[j] * xr[j] - him[j] * xi[j];
            zi[j] = hre[j] * xi[j] + him[j] * xr[j];
        }
        store_cd(bufA, zr, lo, hi);
        store_cd(bufB, zi, lo, hi);

        // inverse (unscaled, G = conj(F)): Wr = fr@Zr + fi@Zi ; Wi = fr@Zi + mfi@Zr
        v8f wr = {}, wi = {};
#pragma unroll
        for (int kb = 0; kb < 4; ++kb) {
            v2f bzr = lds_bfrag(bufA, kb, lo, hi);
            v2f bzi = lds_bfrag(bufB, kb, lo, hi);
            wr = wmma4(fr[kb],  bzr, wr);
            wr = wmma4(fi[kb],  bzi, wr);
            wi = wmma4(fr[kb],  bzi, wi);
            wi = wmma4(mfi[kb], bzr, wi);
        }
        store_cd(bufA, wr, lo, hi);
        store_cd(bufB, wi, lo, hi);
        // y = W*G : yr = Wr@fr + Wi@fi ; yi = Wr@mfi + Wi@fr
        v8f yr = {}, yi = {};
#pragma unroll
        for (int kb = 0; kb < 4; ++kb) {
            v2f awr = lds_afrag(bufA, kb, lo, hi);
            v2f awi = lds_afrag(bufB, kb, lo, hi);
            yr = wmma4(awr, fr[kb],  yr);
            yr = wmma4(awi, fi[kb],  yr);
            yi = wmma4(awr, mfi[kb], yi);
            yi = wmma4(awi, fr[kb],  yi);
        }
        // magnitude (fold 1/256 ifft scale), stage via LDS, b128 store
        v8f mag;
#pragma unroll
        for (int j = 0; j < 8; ++j) {
            float rr = yr[j], ii = yi[j];
            mag[j] = sqrtf(rr * rr + ii * ii) * (1.0f / 256.0f);
        }
        store_cd(bufA, mag, lo, hi);
        float4* ot4 = (float4*)(out + tile * 256);
        float4 o0 = bufA4[rmBase];
        float4 o1 = bufA4[rmBase + 1];
        ot4[lane * 2]     = o0;
        ot4[lane * 2 + 1] = o1;
    }
}

extern "C" void kernel_launch(void* const* d_in, const int* in_sizes, int n_in,
                              void* d_out, int out_size, void* d_ws, size_t ws_size,
                              hipStream_t stream) {
    (void)n_in; (void)out_size; (void)ws_size;
    const float* x     = (const float*)d_in[0];
    const float* filt  = (const float*)d_in[1];
    const float* gamma = (const float*)d_in[2];
    const float* beta  = (const float*)d_in[3];
    float* out = (float*)d_out;

    int n_per_sample = in_sizes[1];          // C*16*16
    int C = n_per_sample / 256;              // 512
    int B = in_sizes[0] / n_per_sample;      // 256
    if (C < 1) C = 1;
    if (B < 1) B = 1;

    float* wsf    = (float*)d_ws;
    float* d_mean = wsf;                     // [B]
    float* d_rstd = wsf + B;                 // [B]
    float* d_Hri  = wsf + 2 * B;             // [C*512] interleaved re/im

    gf_stats<<<B, 256, 0, stream>>>(x, d_mean, d_rstd, n_per_sample);
    gf_filtfft<<<32, 256, 0, stream>>>(filt, d_Hri, C);

    // nwaves = C * wavesPerCh must hold exactly; 8 waves per block.
    int wavesPerCh = 16;                     // 512ch*16 = 8192 waves = 1024 blocks
    while (wavesPerCh > 1 && ((C * wavesPerCh) % 8 != 0 || wavesPerCh > B))
        wavesPerCh >>= 1;
    int blocks = (C * wavesPerCh) / 8;
    if (blocks < 1) blocks = 1;
    gf_main<<<blocks, 256, 0, stream>>>(x, gamma, beta, d_mean, d_rstd,
                                        d_Hri, out, B, C, wavesPerCh);
}